// IndRNN_44229573214895
// MI455X (gfx1250) — compile-verified
//
#include <hip/hip_runtime.h>

typedef float v2f __attribute__((ext_vector_type(2)));
typedef float v8f __attribute__((ext_vector_type(8)));

namespace {

constexpr int kB = 64;    // batch
constexpr int kT = 512;   // timesteps
constexpr int kH = 512;   // hidden
constexpr int kL = 4;     // layers
constexpr int kC = 128;   // head out
constexpr int kLdsStride = 516;  // 512 + 4 pad -> conflict-free column reads

__device__ __forceinline__ v8f wmma_f32(v2f a, v2f b, v8f c) {
  // V_WMMA_F32_16X16X4_F32: D = A(16x4,f32) * B(4x16,f32) + C(16x16,f32)
  return __builtin_amdgcn_wmma_f32_16x16x4_f32(false, a, false, b, (short)0, c,
                                               false, false);
}

}  // namespace

// -------------------------------------------------------------------------
// Input projection: xi[row, n] = sum_k in[row, k] * Wi[n, k] + bi[n] + bh[n]
// rows = B*T flattened. One wave per 32x32 output block (2x2 WMMA tiles):
// each A/B fragment is reused twice -> 4 WMMAs per 4 b64 loads.
// -------------------------------------------------------------------------
__global__ __launch_bounds__(256) void rnn_proj_kernel(
    const float* __restrict__ in, const float* __restrict__ Wi,
    const float* __restrict__ bi, const float* __restrict__ bh,
    float* __restrict__ xi) {
  const int lane   = threadIdx.x & 31;
  const int wave   = threadIdx.x >> 5;
  const int lane15 = lane & 15;
  const int half   = lane >> 4;  // selects K pair per ISA A/B fragment layout

  const int blk   = blockIdx.x * 8 + wave;  // (B*T/32)*(H/32) = 16384 blocks
  const int blkN  = blk & (kH / 32 - 1);    // 0..15
  const int blkM  = blk >> 4;               // 0..1023

  const int n0 = blkN * 32 + lane15;  // col tile 0
  const int n1 = n0 + 16;             // col tile 1
  const int m0 = blkM * 32 + lane15;  // row tile 0 (A row within tile)
  const int m1 = m0 + 16;             // row tile 1

  const float bias0 = bi[n0] + bh[n0];
  const float bias1 = bi[n1] + bh[n1];
  v8f acc00, acc01, acc10, acc11;
#pragma unroll
  for (int r = 0; r < 8; ++r) {
    acc00[r] = bias0; acc01[r] = bias1;
    acc10[r] = bias0; acc11[r] = bias1;
  }

  const float* ap0 = in + (size_t)m0 * kH + half * 2;
  const float* ap1 = in + (size_t)m1 * kH + half * 2;
  const float* bp0 = Wi + (size_t)n0 * kH + half * 2;
  const float* bp1 = Wi + (size_t)n1 * kH + half * 2;
#pragma unroll 4
  for (int ks = 0; ks < kH / 4; ++ks) {
    v2f a0 = *(const v2f*)(ap0 + ks * 4);
    v2f a1 = *(const v2f*)(ap1 + ks * 4);
    v2f b0 = *(const v2f*)(bp0 + ks * 4);
    v2f b1 = *(const v2f*)(bp1 + ks * 4);
    acc00 = wmma_f32(a0, b0, acc00);
    acc01 = wmma_f32(a0, b1, acc01);
    acc10 = wmma_f32(a1, b0, acc10);
    acc11 = wmma_f32(a1, b1, acc11);
  }

  // D layout: row = tileRowBase + half*8 + r, col = n
  float* op0 = xi + (size_t)(blkM * 32 + half * 8) * kH + blkN * 32 + lane15;
  float* op1 = op0 + (size_t)16 * kH;
#pragma unroll
  for (int r = 0; r < 8; ++r) {
    op0[(size_t)r * kH]      = acc00[r];
    op0[(size_t)r * kH + 16] = acc01[r];
    op1[(size_t)r * kH]      = acc10[r];
    op1[(size_t)r * kH + 16] = acc11[r];
  }
}

// -------------------------------------------------------------------------
// Recurrent scan, in-place on xi (buffer layout [B][T][H]):
//   h_t = relu(xi[:, t, :] + h_{t-1} @ Wh^T);  xi[:, t, :] <- h_t
// One workgroup per 16-row batch group; h state lives in LDS; 16 waves each
// own 2 column tiles (32 cols). Step sync = workgroup s_barrier only.
// xi slice for step t+1 is register-prefetched during step t.
// -------------------------------------------------------------------------
__global__ __launch_bounds__(512) void rnn_scan_kernel(
    float* __restrict__ xi, const float* __restrict__ Wh) {
  __shared__ float hbuf[16 * kLdsStride];  // ~32.25 KB

  const int lane   = threadIdx.x & 31;
  const int wave   = threadIdx.x >> 5;  // 0..15
  const int lane15 = lane & 15;
  const int half   = lane >> 4;
  const int bg     = blockIdx.x;  // batch group 0..3 -> rows bg*16..bg*16+15

  for (int i = threadIdx.x; i < 16 * kLdsStride; i += 512) hbuf[i] = 0.0f;

  const int n0 = wave * 32;
  const int nA = n0 + lane15;       // tile 0 column
  const int nB = n0 + 16 + lane15;  // tile 1 column

  const float* aBase = &hbuf[lane15 * kLdsStride + half * 2];
  const float* bpA   = Wh + (size_t)nA * kH + half * 2;
  const float* bpB   = Wh + (size_t)nB * kH + half * 2;

  const size_t rowStride = (size_t)kT * kH;  // stride between batch rows
  float* xBase = xi + (size_t)(bg * 16 + half * 8) * rowStride;
  const int hrow0 = half * 8;

  // Prefetch xi slice for t = 0.
  v8f pfA, pfB;
#pragma unroll
  for (int r = 0; r < 8; ++r) {
    pfA[r] = xBase[(size_t)r * rowStride + nA];
    pfB[r] = xBase[(size_t)r * rowStride + nB];
  }
  __syncthreads();  // hbuf zeroed

  for (int t = 0; t < kT; ++t) {
    v8f accA = pfA;
    v8f accB = pfB;

    // Prefetch xi slice for t+1 (overlaps with this step's WMMAs). Only this
    // wave writes that slice region (at step t+1), so same-wave load->store
    // ordering keeps this race-free.
    if (t + 1 < kT) {
      const float* xn = xBase + (size_t)(t + 1) * kH;
#pragma unroll
      for (int r = 0; r < 8; ++r) {
        pfA[r] = xn[(size_t)r * rowStride + nA];
        pfB[r] = xn[(size_t)r * rowStride + nB];
      }
    }

#pragma unroll 4
    for (int ks = 0; ks < kH / 4; ++ks) {
      v2f a  = *(const v2f*)(aBase + ks * 4);  // shared A frag for both tiles
      v2f b0 = *(const v2f*)(bpA + ks * 4);
      v2f b1 = *(const v2f*)(bpB + ks * 4);
      accA = wmma_f32(a, b0, accA);
      accB = wmma_f32(a, b1, accB);
    }

    __syncthreads();  // all waves done reading h_{t-1} from LDS

    // ReLU + publish h_t to LDS (critical section between the two barriers).
#pragma unroll
    for (int r = 0; r < 8; ++r) {
      accA[r] = accA[r] > 0.0f ? accA[r] : 0.0f;
      accB[r] = accB[r] > 0.0f ? accB[r] : 0.0f;
      hbuf[(hrow0 + r) * kLdsStride + nA] = accA[r];
      hbuf[(hrow0 + r) * kLdsStride + nB] = accB[r];
    }
    __syncthreads();  // h_t visible for next step

    // Layer output store (in-place); overlaps with next step's compute.
    float* xp = xBase + (size_t)t * kH;
#pragma unroll
    for (int r = 0; r < 8; ++r) {
      xp[(size_t)r * rowStride + nA] = accA[r];
      xp[(size_t)r * rowStride + nB] = accB[r];
    }
  }
}

// -------------------------------------------------------------------------
// Head: out[b, c] = sum_k ys[b, T-1, k] * Wfc[c, k] + bfc[c]
// 4 row-tiles x 8 col-tiles = 32 tiles, one wave each.
// -------------------------------------------------------------------------
__global__ __launch_bounds__(256) void head_kernel(
    const float* __restrict__ ys, const float* __restrict__ Wfc,
    const float* __restrict__ bfc, float* __restrict__ out) {
  const int lane   = threadIdx.x & 31;
  const int wave   = threadIdx.x >> 5;
  const int lane15 = lane & 15;
  const int half   = lane >> 4;

  const int tile  = blockIdx.x * 8 + wave;  // 32 tiles
  const int tileN = tile & 7;               // C/16 = 8
  const int tileM = tile >> 3;              // B/16 = 4

  const int n = tileN * 16 + lane15;
  const float biasv = bfc[n];
  v8f acc;
#pragma unroll
  for (int r = 0; r < 8; ++r) acc[r] = biasv;

  const float* ap =
      ys + ((size_t)(tileM * 16 + lane15) * kT + (kT - 1)) * kH + half * 2;
  const float* bp = Wfc + (size_t)n * kH + half * 2;
#pragma unroll 4
  for (int ks = 0; ks < kH / 4; ++ks) {
    v2f a = *(const v2f*)(ap + ks * 4);
    v2f b = *(const v2f*)(bp + ks * 4);
    acc = wmma_f32(a, b, acc);
  }

  float* op = out + (size_t)(tileM * 16 + half * 8) * kC + tileN * 16 + lane15;
#pragma unroll
  for (int r = 0; r < 8; ++r) op[(size_t)r * kC] = acc[r];
}

// -------------------------------------------------------------------------
extern "C" void kernel_launch(void* const* d_in, const int* in_sizes, int n_in,
                              void* d_out, int out_size, void* d_ws,
                              size_t ws_size, hipStream_t stream) {
  const float* x    = (const float*)d_in[0];  // [B,T,H]
  const float* W_ih = (const float*)d_in[1];  // [L,H,H]
  const float* W_hh = (const float*)d_in[2];  // [L,H,H]
  const float* b_ih = (const float*)d_in[3];  // [L,H]
  const float* b_hh = (const float*)d_in[4];  // [L,H]
  const float* W_fc = (const float*)d_in[5];  // [C,H]
  const float* b_fc = (const float*)d_in[6];  // [C]

  // Workspace: two ping-pong activation buffers of B*T*H floats (64 MB each).
  float* bufA = (float*)d_ws;
  float* bufB = bufA + (size_t)kB * kT * kH;

  const int projWaveBlks = (kB * kT / 32) * (kH / 32);  // 16384 (32x32 blocks)
  const int projBlocks   = projWaveBlks / 8;            // 2048 CTAs x 8 waves

  const float* cur = x;
  float* dst = bufA;
  float* alt = bufB;
  for (int l = 0; l < kL; ++l) {
    rnn_proj_kernel<<<projBlocks, 256, 0, stream>>>(
        cur, W_ih + (size_t)l * kH * kH, b_ih + (size_t)l * kH,
        b_hh + (size_t)l * kH, dst);
    rnn_scan_kernel<<<4, 512, 0, stream>>>(dst, W_hh + (size_t)l * kH * kH);
    cur = dst;
    float* tmp = dst;
    dst = alt;
    alt = tmp;
  }
  head_kernel<<<4, 256, 0, stream>>>(cur, W_fc, b_fc, (float*)d_out);

  (void)in_sizes; (void)n_in; (void)out_size; (void)ws_size;
}